// LinearRNN_11072425689983
// MI455X (gfx1250) — compile-verified
//
#include <hip/hip_runtime.h>

// Problem constants (match reference)
#define B_    64
#define T_    2048
#define DIN   512
#define DH    1024
#define KTOT  (DH + DIN)        // 1536: [h | x] fused K
#define NWG   64                // workgroups; each owns 16 output columns
#define CPW   16                // columns per workgroup
#define THREADS 128             // 4 waves (wave32): wave w = M-tile w (rows 16w..16w+15)
#define LDS_STRIDE 1540         // 1536 + 4 pad -> 16-lane reads hit distinct banks

typedef __attribute__((ext_vector_type(2))) float v2f;
typedef __attribute__((ext_vector_type(8))) float v8f;

// ---- grid-wide barrier (sense-reversing, state in d_ws) -------------------
__device__ __forceinline__ void grid_sync(unsigned* bar, unsigned nwg) {
    __syncthreads();
    if (threadIdx.x == 0) {
        __threadfence();
        unsigned g = __hip_atomic_load(&bar[1], __ATOMIC_RELAXED, __HIP_MEMORY_SCOPE_AGENT);
        if (atomicAdd(&bar[0], 1u) == nwg - 1u) {
            __hip_atomic_store(&bar[0], 0u, __ATOMIC_RELAXED, __HIP_MEMORY_SCOPE_AGENT);
            __threadfence();
            atomicAdd(&bar[1], 1u);
        } else {
            while (__hip_atomic_load(&bar[1], __ATOMIC_RELAXED, __HIP_MEMORY_SCOPE_AGENT) == g)
                __builtin_amdgcn_s_sleep(2);
        }
        __threadfence();
    }
    __syncthreads();
}

// ---- init: zero h ping buffer 0 + barrier words (d_ws is poisoned 0xAA) ---
__global__ void rnn_init(float* ws) {
    int i = blockIdx.x * blockDim.x + threadIdx.x;
    if (i < B_ * DH) ws[i] = 0.0f;
    if (i == 0) {
        unsigned* bar = (unsigned*)(ws + 2 * B_ * DH);
        bar[0] = 0u; bar[1] = 0u;
    }
}

// ---- persistent recurrence kernel -----------------------------------------
__global__ void __launch_bounds__(THREADS)
rnn_persistent(const float* __restrict__ data,
               const int*   __restrict__ batch_sizes,
               const int*   __restrict__ unsort,
               const float* __restrict__ W_ih,
               const float* __restrict__ b_ih,
               const float* __restrict__ W_hh,
               const float* __restrict__ b_hh,
               float*       __restrict__ out,
               float*       __restrict__ ws)
{
    extern __shared__ float w_lds[];            // [CPW][LDS_STRIDE] fused [W_hh|W_ih] slice
    float* h0 = ws;                             // ping
    float* h1 = ws + B_ * DH;                   // pong
    unsigned* bar = (unsigned*)(ws + 2 * B_ * DH);

    const int tid  = threadIdx.x;
    const int wg   = blockIdx.x;
    const int lane = tid & 31;
    const int wave = tid >> 5;                  // 0..3 -> M-tile
    const int l15  = lane & 15;
    const int hi   = lane >> 4;                 // lane half selects K pair / M half
    const int row0 = wave * 16;
    const int n0   = wg * CPW;                  // this WG's output columns [n0, n0+16)

    // Stage weight slice into LDS: w_lds[c][k] = k<DH ? W_hh[n0+c][k] : W_ih[n0+c][k-DH]
    for (int idx = tid; idx < CPW * KTOT; idx += THREADS) {
        int c = idx / KTOT;
        int k = idx - c * KTOT;
        float v = (k < DH) ? W_hh[(size_t)(n0 + c) * DH + k]
                           : W_ih[(size_t)(n0 + c) * DIN + (k - DH)];
        w_lds[c * LDS_STRIDE + k] = v;
    }
    __syncthreads();

    const int   gcol  = n0 + l15;               // this lane's output column
    const float bsum  = b_ih[gcol] + b_hh[gcol];
    const int   arow  = row0 + l15;             // this lane's A row (batch row)
    const int   bbase = l15 * LDS_STRIDE + 2 * hi;

    int offset = 0;                             // packed-data row offset (prefix of batch_sizes)
    for (int t = 0; t < T_; ++t) {
        const int bs = batch_sizes[t];
        const float* hin  = (t & 1) ? h1 : h0;
        float*       hout = (t & 1) ? h0 : h1;

        v8f acc = {0.f, 0.f, 0.f, 0.f, 0.f, 0.f, 0.f, 0.f};

        // ---- h part: K = 0..DH, A = h tile (16x4 f32 per ISA layout) ----
        const float* aptr = hin + (size_t)arow * DH + 2 * hi;
        #pragma unroll 4
        for (int kk = 0; kk < DH; kk += 4) {
            v2f a = *(const v2f*)(aptr + kk);
            v2f b = *(const v2f*)(&w_lds[bbase + kk]);
            acc = __builtin_amdgcn_wmma_f32_16x16x4_f32(
                      false, a, false, b, (short)0, acc, false, false);
        }
        // ---- x part: K = DH..KTOT, A = packed input rows (clamped for frozen rows)
        const int rc = (arow < bs) ? arow : (bs - 1);
        const float* xptr = data + (size_t)(offset + rc) * DIN + 2 * hi;
        #pragma unroll 4
        for (int kk = 0; kk < DIN; kk += 4) {
            v2f a = *(const v2f*)(xptr + kk);
            v2f b = *(const v2f*)(&w_lds[bbase + DH + kk]);
            acc = __builtin_amdgcn_wmma_f32_16x16x4_f32(
                      false, a, false, b, (short)0, acc, false, false);
        }

        // Hint next step's input row into cache (global_prefetch_b8)
        if (t + 1 < T_)
            __builtin_prefetch(data + (size_t)(offset + bs + rc) * DIN, 0, 1);

        // ---- store C/D (16x16 f32: VGPR r, lanes -> row r+8*hi, col l15);
        //      rows >= bs keep previous value (freeze semantics)
        const int rbase = row0 + 8 * hi;
        #pragma unroll
        for (int r = 0; r < 8; ++r) {
            int row = rbase + r;
            float v = (row < bs) ? (acc[r] + bsum)
                                 : hin[(size_t)row * DH + gcol];
            hout[(size_t)row * DH + gcol] = v;
        }

        offset += bs;
        grid_sync(bar, NWG);
    }

    // ---- epilogue: T_ even -> final state in h0; apply unsort permutation.
    // Each WG writes only its own 16 columns (which it produced itself).
    for (int e = tid; e < B_ * CPW; e += THREADS) {
        int j = e >> 4;                 // output batch row
        int c = e & (CPW - 1);
        int col = n0 + c;
        out[(size_t)j * DH + col] = h0[(size_t)unsort[j] * DH + col];
    }
}

extern "C" void kernel_launch(void* const* d_in, const int* in_sizes, int n_in,
                              void* d_out, int out_size, void* d_ws, size_t ws_size,
                              hipStream_t stream) {
    (void)in_sizes; (void)n_in; (void)out_size; (void)ws_size;
    const float* data        = (const float*)d_in[0];
    const int*   batch_sizes = (const int*)  d_in[1];
    const int*   unsort      = (const int*)  d_in[2];
    const float* W_ih        = (const float*)d_in[3];
    const float* b_ih        = (const float*)d_in[4];
    const float* W_hh        = (const float*)d_in[5];
    const float* b_hh        = (const float*)d_in[6];
    float* out = (float*)d_out;
    float* ws  = (float*)d_ws;   // needs 2*64*1024 floats + 2 uints = 512 KB + 8 B

    rnn_init<<<(B_ * DH + 255) / 256, 256, 0, stream>>>(ws);

    size_t lds_bytes = (size_t)CPW * LDS_STRIDE * sizeof(float);   // ~96 KB of 320 KB/WGP
    rnn_persistent<<<NWG, THREADS, lds_bytes, stream>>>(
        data, batch_sizes, unsort, W_ih, b_ih, W_hh, b_hh, out, ws);
}